// graphnetwork_1529008357904
// MI455X (gfx1250) — compile-verified
//
#include <hip/hip_runtime.h>

typedef __attribute__((ext_vector_type(2))) float v2f;
typedef __attribute__((ext_vector_type(8))) float v8f;

#define N_NODES 50000
#define N_EDGES 400000
#define NB      2
#define M_ROWS  (N_NODES * NB)      // 100000 rows, 6250 tiles of 16
#define N_TILES (M_ROWS / 16)

// ---------------- degree -> 1/max(deg,1) ----------------
__global__ void deg_count_kernel(const int* __restrict__ dst, float* __restrict__ deg) {
    int e = blockIdx.x * blockDim.x + threadIdx.x;
    if (e < N_EDGES) atomicAdd(&deg[dst[e]], 1.0f);
}

__global__ void deg_finalize_kernel(float* __restrict__ deg) {
    int n = blockIdx.x * blockDim.x + threadIdx.x;
    if (n < N_NODES) deg[n] = 1.0f / fmaxf(deg[n], 1.0f);
}

// ---------------- in_feat [B,N,64] -> act [node*2+b, 64] ----------------
__global__ void transpose_in_kernel(const float* __restrict__ in_feat, float* __restrict__ out) {
    int idx = blockIdx.x * blockDim.x + threadIdx.x;   // over M_ROWS*64
    if (idx >= M_ROWS * 64) return;
    int f = idx & 63;
    int m = idx >> 6;
    int node = m >> 1, b = m & 1;
    out[idx] = in_feat[(b * N_NODES + node) * 64 + f];
}

// ---------------- edge scatter-add: agg[dst] += x[src] ----------------
// x, agg layout: [node*2+b, C] dense => 2*C contiguous floats per node.
__global__ void scatter_kernel(const float* __restrict__ x, const int* __restrict__ src,
                               const int* __restrict__ dst, float* __restrict__ agg, int C2) {
    const int chunks = C2 >> 2;                        // float4 chunks per node-pair row
    int tid = blockIdx.x * blockDim.x + threadIdx.x;
    int total = N_EDGES * chunks;
    if (tid >= total) return;
    int e  = tid / chunks;
    int c4 = (tid - e * chunks) << 2;
    int s = src[e], d = dst[e];
    const float4 v = *(const float4*)(x + s * C2 + c4);
    float* a = agg + d * C2 + c4;
    atomicAdd(a + 0, v.x);
    atomicAdd(a + 1, v.y);
    atomicAdd(a + 2, v.z);
    atomicAdd(a + 3, v.w);
}

// ---------------- fused SAGE GEMM: OUT = act(X@Ws + (AGG*invdeg)@Wn + b) ----------------
// One wave32 per 16-row tile. WMMA f32 16x16x4, K stepped by 4, two WMMAs per
// k-step (self + neighbor) accumulated into the same C tile.
__global__ void sage_gemm_kernel(const float* __restrict__ X, const float* __restrict__ AGG,
                                 const float* __restrict__ Ws, const float* __restrict__ Wn,
                                 const float* __restrict__ bias, const float* __restrict__ invdeg,
                                 float* __restrict__ OUT,
                                 int Fin, int Fout, int ldo, int relu, int transpose_out) {
    const int lane = threadIdx.x & 31;
    const int wave = threadIdx.x >> 5;
    const int wavesPerBlock = blockDim.x >> 5;
    const int tile = blockIdx.x * wavesPerBlock + wave;
    if (tile >= N_TILES) return;                       // wave-uniform: EXEC stays all-1s

    const int hi  = lane >> 4;                         // 0: k even pair low, 1: high
    const int col = lane & 15;                         // A row index == B/N col index
    const int rowA = tile * 16 + col;                  // lane's A-matrix row
    const float sA = invdeg[rowA >> 1];                // per-row mean scale (node = m/2)
    const float* __restrict__ xrow = X   + rowA * Fin;
    const float* __restrict__ grow = AGG + rowA * Fin;

    for (int nt = 0; nt < Fout; nt += 16) {
        v8f acc = {};
        for (int k = 0; k < Fin; k += 4) {
            const int kk = k + 2 * hi;                 // ISA 7.12.2: A vgpr j holds K=kk+j
            v2f a = *(const v2f*)(xrow + kk);
            v2f g = *(const v2f*)(grow + kk);
            g = g * sA;                                // fold 1/deg into neighbor A tile
            v2f bs, bn;                                // B vgpr j holds K=kk+j at N=col
            bs.x = Ws[(kk    ) * Fout + nt + col];
            bs.y = Ws[(kk + 1) * Fout + nt + col];
            bn.x = Wn[(kk    ) * Fout + nt + col];
            bn.y = Wn[(kk + 1) * Fout + nt + col];
            acc = __builtin_amdgcn_wmma_f32_16x16x4_f32(false, a, false, bs, (short)0, acc, false, false);
            acc = __builtin_amdgcn_wmma_f32_16x16x4_f32(false, g, false, bn, (short)0, acc, false, false);
        }
        const float bv = bias[nt + col];
#pragma unroll
        for (int r = 0; r < 8; ++r) {                  // D vgpr r -> row r+8*hi, col=lane&15
            int m = tile * 16 + r + 8 * hi;
            float v = acc[r] + bv;
            if (relu) v = fmaxf(v, 0.0f);
            if (transpose_out) {                       // write [B,N,64] final layout
                int node = m >> 1, bb = m & 1;
                OUT[(bb * N_NODES + node) * 64 + nt + col] = v;
            } else {
                OUT[m * ldo + nt + col] = v;
            }
        }
    }
}

// ---------------- concat tail: OUT[:, chanbase:chanbase+C] = X ----------------
__global__ void concat_copy_kernel(const float* __restrict__ X, float* __restrict__ OUT,
                                   int C, int ldo, int chanbase) {
    int idx = blockIdx.x * blockDim.x + threadIdx.x;   // over M_ROWS*C
    if (idx >= M_ROWS * C) return;
    int m = idx / C;
    int c = idx - m * C;
    OUT[m * ldo + chanbase + c] = X[idx];
}

extern "C" void kernel_launch(void* const* d_in, const int* in_sizes, int n_in,
                              void* d_out, int out_size, void* d_ws, size_t ws_size,
                              hipStream_t stream) {
    (void)in_sizes; (void)n_in; (void)out_size; (void)ws_size;

    const float* in_feat = (const float*)d_in[0];
    const int*   src     = (const int*)d_in[1];
    const int*   dst     = (const int*)d_in[2];
    // per-layer params: d_in[3 + 3*(i-1)] = w_self, +1 = w_neigh, +2 = bias
    const float* WS[6]; const float* WN[6]; const float* BI[6];
    for (int i = 0; i < 6; ++i) {
        WS[i] = (const float*)d_in[3 + 3 * i];
        WN[i] = (const float*)d_in[4 + 3 * i];
        BI[i] = (const float*)d_in[5 + 3 * i];
    }
    static const int FIN[6]  = {64, 128, 64, 32, 64, 128};
    static const int FOUT[6] = {128, 64, 32, 32, 64, 64};

    // workspace layout (floats)
    const size_t BUF = (size_t)M_ROWS * 128;           // 12.8M floats per buffer
    float* invdeg = (float*)d_ws;
    float* act0 = invdeg + 51200;
    float* act1 = act0 + BUF;
    float* act2 = act1 + BUF;
    float* agg  = act2 + BUF;

    const int T = 256;

    // 1/deg
    hipMemsetAsync(invdeg, 0, N_NODES * sizeof(float), stream);
    deg_count_kernel<<<(N_EDGES + T - 1) / T, T, 0, stream>>>(dst, invdeg);
    deg_finalize_kernel<<<(N_NODES + T - 1) / T, T, 0, stream>>>(invdeg);

    // [B,N,64] -> [m,64]
    transpose_in_kernel<<<(M_ROWS * 64 + T - 1) / T, T, 0, stream>>>(in_feat, act0);

    // run one sage layer: scatter + fused gemm
    auto sage = [&](int li, const float* x, float* out, int ldo, int relu, int tro) {
        const int Fin = FIN[li], Fout = FOUT[li];
        const int C2 = 2 * Fin;
        hipMemsetAsync(agg, 0, (size_t)M_ROWS * Fin * sizeof(float), stream);
        int scat_total = N_EDGES * (C2 >> 2);
        scatter_kernel<<<(scat_total + T - 1) / T, T, 0, stream>>>(x, src, dst, agg, C2);
        const int wavesPerBlock = 4;                   // 128 threads = 4 wave32
        int grid = (N_TILES + wavesPerBlock - 1) / wavesPerBlock;
        sage_gemm_kernel<<<grid, wavesPerBlock * 32, 0, stream>>>(
            x, agg, WS[li], WN[li], BI[li], invdeg, out, Fin, Fout, ldo, relu, tro);
    };

    for (int step = 0; step < 2; ++step) {
        // L1: act0(64) -> act1(128), relu
        sage(0, act0, act1, 128, 1, 0);
        // L2: act1(128) -> act2(64), relu
        sage(1, act1, act2, 64, 1, 0);
        // L3: act2(64) -> act0(32), relu
        sage(2, act2, act0, 32, 1, 0);
        // L4: act0(32) -> act1[0:32] relu, concat act0 -> act1[32:64]
        sage(3, act0, act1, 64, 1, 0);
        concat_copy_kernel<<<(M_ROWS * 32 + T - 1) / T, T, 0, stream>>>(act0, act1, 32, 64, 32);
        // L5: act1(64) -> act2[0:64] relu, concat act1 -> act2[64:128]
        sage(4, act1, act2, 128, 1, 0);
        concat_copy_kernel<<<(M_ROWS * 64 + T - 1) / T, T, 0, stream>>>(act1, act2, 64, 128, 64);
        // L6: act2(128) -> act0(64) no relu; last step writes d_out transposed
        if (step == 1) {
            sage(5, act2, (float*)d_out, 64, 0, 1);
        } else {
            sage(5, act2, act0, 64, 0, 0);
        }
    }
}